// ProposalTargetAssigner_46548855554068
// MI455X (gfx1250) — compile-verified
//
#include <hip/hip_runtime.h>
#include <hip/hip_bf16.h>
#include <math.h>

// ---------------------------------------------------------------------------
// ProposalTargetAssigner for MI455X (gfx1250, wave32)
//   boxes   : [32][7]  float32
//   anchors : [3][2][160][160][7] float32
//   class_idx: [32] int32
// Outputs (concatenated float32):
//   G_cls [3][2][160][160]       (153600)
//   G_reg [3][2][160][160][7]    (1075200)
//   M_cls [1][2][160][160]       (51200)
//   M_reg [3][1][160][160][1]    (76800)
// ---------------------------------------------------------------------------

#define N_CLS   3
#define NPOS    51200            // 2*160*160 positions per class
#define NPOS0   25600            // yaw==0 positions
#define NB      32
#define IOU_EPS 1e-8f

// workspace layout (float units)
#define WS_HIGH_OFF (N_CLS * NB * NPOS)   // iou matrix first, then 96 "highest" slots

// output offsets (float units)
#define OFF_GCLS 0
#define OFF_GREG (N_CLS * NPOS)                 // 153600
#define OFF_MCLS (OFF_GREG + N_CLS * NPOS * 7)  // 1228800
#define OFF_MREG (OFF_MCLS + NPOS)              // 1280000

typedef float v2f __attribute__((ext_vector_type(2)));
typedef float v8f __attribute__((ext_vector_type(8)));

// ---------------------------------------------------------------------------
// Exact rotated-rectangle IoU, mirroring the reference _pair_iou semantics.
// b1 = ground-truth box, b2 = anchor (bev: x,y,w,l,yaw with cos/sin given).
// ---------------------------------------------------------------------------
__device__ __noinline__ float pairIoU(float x1, float y1, float w1, float l1,
                                      float cs1, float sn1,
                                      float x2, float y2, float w2, float l2,
                                      float cs2, float sn2) {
  float C1x[4], C1y[4], C2x[4], C2y[4];
  {
    float ux = cs1 * (w1 * 0.5f), uy = sn1 * (w1 * 0.5f);
    float vx = -sn1 * (l1 * 0.5f), vy = cs1 * (l1 * 0.5f);
    C1x[0] = x1 + ux + vx; C1y[0] = y1 + uy + vy;
    C1x[1] = x1 + ux - vx; C1y[1] = y1 + uy - vy;
    C1x[2] = x1 - ux - vx; C1y[2] = y1 - uy - vy;
    C1x[3] = x1 - ux + vx; C1y[3] = y1 - uy + vy;
  }
  {
    float ux = cs2 * (w2 * 0.5f), uy = sn2 * (w2 * 0.5f);
    float vx = -sn2 * (l2 * 0.5f), vy = cs2 * (l2 * 0.5f);
    C2x[0] = x2 + ux + vx; C2y[0] = y2 + uy + vy;
    C2x[1] = x2 + ux - vx; C2y[1] = y2 + uy - vy;
    C2x[2] = x2 - ux - vx; C2y[2] = y2 - uy - vy;
    C2x[3] = x2 - ux + vx; C2y[3] = y2 - uy + vy;
  }
  float px[24], py[24];
  int nv = 0;
  // corners of box inside anchor
#pragma unroll
  for (int i = 0; i < 4; ++i) {
    float rx = C1x[i] - x2, ry = C1y[i] - y2;
    float du = rx * cs2 + ry * sn2;
    float dv = -rx * sn2 + ry * cs2;
    if (fabsf(du) <= w2 * 0.5f + 1e-5f && fabsf(dv) <= l2 * 0.5f + 1e-5f) {
      px[nv] = C1x[i]; py[nv] = C1y[i]; ++nv;
    }
  }
  // corners of anchor inside box
#pragma unroll
  for (int i = 0; i < 4; ++i) {
    float rx = C2x[i] - x1, ry = C2y[i] - y1;
    float du = rx * cs1 + ry * sn1;
    float dv = -rx * sn1 + ry * cs1;
    if (fabsf(du) <= w1 * 0.5f + 1e-5f && fabsf(dv) <= l1 * 0.5f + 1e-5f) {
      px[nv] = C2x[i]; py[nv] = C2y[i]; ++nv;
    }
  }
  // edge-edge intersections (16 pairs)
#pragma unroll
  for (int i = 0; i < 4; ++i) {
    float d1x = C1x[(i + 1) & 3] - C1x[i];
    float d1y = C1y[(i + 1) & 3] - C1y[i];
#pragma unroll
    for (int j = 0; j < 4; ++j) {
      float d2x = C2x[(j + 1) & 3] - C2x[j];
      float d2y = C2y[(j + 1) & 3] - C2y[j];
      float den = d1x * d2y - d1y * d2x;
      if (fabsf(den) >= IOU_EPS) {
        float qpx = C2x[j] - C1x[i], qpy = C2y[j] - C1y[i];
        float t = (qpx * d2y - qpy * d2x) / den;
        float s = (qpx * d1y - qpy * d1x) / den;
        if (t >= 0.f && t <= 1.f && s >= 0.f && s <= 1.f) {
          px[nv] = C1x[i] + t * d1x;
          py[nv] = C1y[i] + t * d1y;
          ++nv;
        }
      }
    }
  }
  if (nv == 0) return 0.f;
  float cx = 0.f, cy = 0.f;
  for (int k = 0; k < nv; ++k) { cx += px[k]; cy += py[k]; }
  float inv = 1.f / (float)nv;
  cx *= inv; cy *= inv;
  float ang[24];
  for (int k = 0; k < nv; ++k) ang[k] = atan2f(py[k] - cy, px[k] - cx);
  // stable insertion sort by angle (matches stable argsort tie-break)
  for (int k = 1; k < nv; ++k) {
    float ka = ang[k], kx = px[k], ky = py[k];
    int j = k - 1;
    while (j >= 0 && ang[j] > ka) {
      ang[j + 1] = ang[j]; px[j + 1] = px[j]; py[j + 1] = py[j]; --j;
    }
    ang[j + 1] = ka; px[j + 1] = kx; py[j + 1] = ky;
  }
  float a2 = 0.f;
  for (int k = 0; k < nv; ++k) {
    int n = (k + 1 == nv) ? 0 : k + 1;
    a2 += px[k] * py[n] - px[n] * py[k];
  }
  float inter = 0.5f * fabsf(a2);
  float uni = w1 * l1 + w2 * l2 - inter;
  float iou = inter / fmaxf(uni, IOU_EPS);
  return fminf(fmaxf(iou, 0.f), 1.f);
}

// ---------------------------------------------------------------------------
__global__ void pta_init_kernel(unsigned* highest) {
  int t = threadIdx.x;
  if (t < N_CLS * NB) highest[t] = 0u;  // bits of +0.0f
}

// ---------------------------------------------------------------------------
// Pass A: wave = 32 consecutive anchors of one class.
// WMMA 16x16x4 F32 computes the 32x32 center-distance^2 tile used for the
// circumradius cull; exact IoU only for surviving (valid, close) pairs.
// ---------------------------------------------------------------------------
__global__ void __launch_bounds__(256)
pta_iou_kernel(const float* __restrict__ boxes,
               const float* __restrict__ anchors,
               const int* __restrict__ class_idx,
               float* __restrict__ ws_iou,
               unsigned* __restrict__ ws_high) {
  // per-box data: 0:x 1:y 2:w 3:l 4:cos 5:sin 6:|c|^2 7:circumradius
  __shared__ float sb[NB][8];
  __shared__ unsigned s_valid[N_CLS];
  int tid = threadIdx.x;
  if (tid < NB) {
    const float* B = boxes + tid * 7;
    float x = B[0], y = B[1], w = B[3], l = B[4], yaw = B[6];
    sb[tid][0] = x; sb[tid][1] = y; sb[tid][2] = w; sb[tid][3] = l;
    sb[tid][4] = cosf(yaw); sb[tid][5] = sinf(yaw);
    sb[tid][6] = x * x + y * y;
    sb[tid][7] = 0.5f * sqrtf(w * w + l * l);
  }
  if (tid < N_CLS) {
    unsigned m = 0;
    for (int b = 0; b < NB; ++b)
      if (class_idx[b] == tid) m |= (1u << b);
    s_valid[tid] = m;
  }
  __syncthreads();

  int lane = tid & 31;
  int wvid = blockIdx.x * (blockDim.x >> 5) + (tid >> 5);
  int c = wvid / (NPOS / 32);             // class: 1600 waves per class
  int p = (wvid % (NPOS / 32)) * 32 + lane;

  const float* A = anchors + ((size_t)c * NPOS + p) * 7;
  float ax = A[0], ay = A[1], aw = A[3], al = A[4], ayaw = A[6];
  float acs = cosf(ayaw), asn = sinf(ayaw);
  float aq = ax * ax + ay * ay;
  float rA = 0.5f * sqrtf(aw * aw + al * al);  // uniform within a class

  // ---- WMMA distance cull: D = A(16x4) x B(4x16), dist^2 tile ----
  bool lo = lane < 16;
  int nn = lane & 15;
  // A tile rows via cross-lane gather (all lanes active)
  float ax0 = __shfl(ax, nn, 32),      ay0 = __shfl(ay, nn, 32),      aq0 = __shfl(aq, nn, 32);
  float ax1 = __shfl(ax, nn + 16, 32), ay1 = __shfl(ay, nn + 16, 32), aq1 = __shfl(aq, nn + 16, 32);
  v2f Am0, Am1;
  Am0.x = lo ? ax0 : aq0;  Am0.y = lo ? ay0 : 1.0f;   // K0/K2, K1/K3
  Am1.x = lo ? ax1 : aq1;  Am1.y = lo ? ay1 : 1.0f;
  v2f Bm0, Bm1;
  {
    float bx0 = sb[nn][0],      by0 = sb[nn][1],      bq0 = sb[nn][6];
    float bx1 = sb[nn + 16][0], by1 = sb[nn + 16][1], bq1 = sb[nn + 16][6];
    Bm0.x = lo ? (-2.f * bx0) : 1.0f;  Bm0.y = lo ? (-2.f * by0) : bq0;
    Bm1.x = lo ? (-2.f * bx1) : 1.0f;  Bm1.y = lo ? (-2.f * by1) : bq1;
  }
  v8f cz = {0.f, 0.f, 0.f, 0.f, 0.f, 0.f, 0.f, 0.f};
  v8f D00 = __builtin_amdgcn_wmma_f32_16x16x4_f32(false, Am0, false, Bm0, (short)0, cz, false, false);
  v8f D01 = __builtin_amdgcn_wmma_f32_16x16x4_f32(false, Am0, false, Bm1, (short)0, cz, false, false);
  v8f D10 = __builtin_amdgcn_wmma_f32_16x16x4_f32(false, Am1, false, Bm0, (short)0, cz, false, false);
  v8f D11 = __builtin_amdgcn_wmma_f32_16x16x4_f32(false, Am1, false, Bm1, (short)0, cz, false, false);

  // thresholds per lane correspond to box N = lane&15 of each group
  float t0 = rA + sb[nn][7] + 0.01f;       t0 *= t0;
  float t1 = rA + sb[nn + 16][7] + 0.01f;  t1 *= t1;

  unsigned bal[2][2][8];  // [anchorTile][boxGroup][vgpr]
#pragma unroll
  for (int v = 0; v < 8; ++v) {
    bal[0][0][v] = __builtin_amdgcn_ballot_w32(D00[v] <= t0);
    bal[0][1][v] = __builtin_amdgcn_ballot_w32(D01[v] <= t1);
    bal[1][0][v] = __builtin_amdgcn_ballot_w32(D10[v] <= t0);
    bal[1][1][v] = __builtin_amdgcn_ballot_w32(D11[v] <= t1);
  }
  // my anchor = wave-anchor `lane`: tile = lane>>4, row m = lane&15.
  // D layout: lane l -> N=l&15, VGPR v -> M = v + 8*(l>=16).
  int myt = lane >> 4;
  int m = nn;
  unsigned e0 = bal[myt][0][m & 7];
  unsigned e1 = bal[myt][1][m & 7];
  unsigned g0 = (m < 8) ? (e0 & 0xFFFFu) : (e0 >> 16);
  unsigned g1 = (m < 8) ? (e1 & 0xFFFFu) : (e1 >> 16);
  unsigned cull = g0 | (g1 << 16);
  unsigned vmask = s_valid[c];
  unsigned work = cull & vmask;

  // ---- exact IoU for surviving pairs; stream full masked matrix ----
#pragma unroll 1
  for (int b = 0; b < NB; ++b) {
    float v;
    if (!((vmask >> b) & 1u)) {
      v = -1.0f;                       // invalid class -> masked
    } else if (!((work >> b) & 1u)) {
      v = 0.0f;                        // culled: provably zero overlap
    } else {
      v = pairIoU(sb[b][0], sb[b][1], sb[b][2], sb[b][3], sb[b][4], sb[b][5],
                  ax, ay, aw, al, acs, asn);
    }
    ws_iou[((size_t)(c * NB + b)) * NPOS + p] = v;
    if (v > 0.f)
      atomicMax(&ws_high[c * NB + b], __float_as_uint(v));  // positive floats: bit order == value order
  }
}

// ---------------------------------------------------------------------------
// Pass B: one thread per (yaw,y,x) position; per-class match + labels,
// cross-class resolution, regression encoding, all outputs.
// ---------------------------------------------------------------------------
__global__ void __launch_bounds__(256)
pta_assign_kernel(const float* __restrict__ boxes,
                  const float* __restrict__ anchors,
                  const float* __restrict__ ws_iou,
                  const unsigned* __restrict__ ws_high,
                  float* __restrict__ out) {
  int p = blockIdx.x * blockDim.x + threadIdx.x;
  if (p >= NPOS) return;
  const float LOWT[N_CLS]  = {0.45f, 0.35f, 0.35f};
  const float HIGHT[N_CLS] = {0.60f, 0.50f, 0.50f};

  int lab[N_CLS], arg[N_CLS];
#pragma unroll
  for (int c = 0; c < N_CLS; ++c) {
    float mv = -2.0f;
    int am = 0;
    bool best = false;
    for (int b = 0; b < NB; ++b) {
      float v = ws_iou[((size_t)(c * NB + b)) * NPOS + p];
      if (v > mv) { mv = v; am = b; }   // strict > : first-index tie-break
      if (v > 0.f && __float_as_uint(v) == ws_high[c * NB + b]) best = true;
    }
    int L = (mv >= HIGHT[c]) ? 1 : ((mv >= LOWT[c]) ? -1 : 0);
    if (best) L = 1;                    // ALLOW_LOW_QUALITY override
    lab[c] = L;
    arg[c] = am;
  }

  // cross-class resolution (mirrors reference ordering)
  int npos = (lab[0] == 1) + (lab[1] == 1) + (lab[2] == 1);
  if (npos > 1) { lab[0] = lab[1] = lab[2] = -1; }
  bool negative = (lab[0] == 0) || (lab[1] == 0) || (lab[2] == 0);
  int npos2 = (lab[0] == 1) + (lab[1] == 1) + (lab[2] == 1);
  bool positive = (npos2 == 1);
  if (negative && !positive) { lab[0] = lab[1] = lab[2] = 0; }
  bool loss_mask = !((lab[0] == -1) && (lab[1] == -1) && (lab[2] == -1));
#pragma unroll
  for (int c = 0; c < N_CLS; ++c)
    if (lab[c] == -1) lab[c] = 0;

  // outputs
  out[OFF_MCLS + p] = loss_mask ? 1.0f : 0.0f;
#pragma unroll
  for (int c = 0; c < N_CLS; ++c) {
    float lf = (float)lab[c];
    out[OFF_GCLS + (size_t)c * NPOS + p] = lf;
    if (p < NPOS0) out[OFF_MREG + (size_t)c * NPOS0 + p] = lf;
    size_t rbase = OFF_GREG + ((size_t)c * NPOS + p) * 7;
    if (lab[c] == 1) {
      const float* G = boxes + arg[c] * 7;
      const float* A = anchors + ((size_t)c * NPOS + p) * 7;
      float nrm = sqrtf(A[3] * A[3] + A[4] * A[4]);
      out[rbase + 0] = (G[0] - A[0]) / nrm;
      out[rbase + 1] = (G[1] - A[1]) / nrm;
      out[rbase + 2] = (G[2] - A[2]) / A[5];
      out[rbase + 3] = logf(G[3] / A[3]);
      out[rbase + 4] = logf(G[4] / A[4]);
      out[rbase + 5] = logf(G[5] / A[5]);
      out[rbase + 6] = G[6] - A[6];
    } else {
#pragma unroll
      for (int k = 0; k < 7; ++k) out[rbase + k] = 0.0f;
    }
  }
}

// ---------------------------------------------------------------------------
extern "C" void kernel_launch(void* const* d_in, const int* in_sizes, int n_in,
                              void* d_out, int out_size, void* d_ws, size_t ws_size,
                              hipStream_t stream) {
  (void)in_sizes; (void)n_in; (void)out_size; (void)ws_size;
  const float* boxes   = (const float*)d_in[0];
  const float* anchors = (const float*)d_in[1];
  const int* class_idx = (const int*)d_in[2];
  float* out = (float*)d_out;
  float* ws_iou = (float*)d_ws;
  unsigned* ws_high = (unsigned*)(ws_iou + WS_HIGH_OFF);

  pta_init_kernel<<<1, 128, 0, stream>>>(ws_high);
  // 3*51200 anchors / 32 per wave = 4800 waves = 600 blocks of 8 waves
  pta_iou_kernel<<<600, 256, 0, stream>>>(boxes, anchors, class_idx, ws_iou, ws_high);
  // 51200 positions
  pta_assign_kernel<<<200, 256, 0, stream>>>(boxes, anchors, ws_iou, ws_high, out);
}